// TanhFixedPointLayer_62130996903974
// MI455X (gfx1250) — compile-verified
//
#include <hip/hip_runtime.h>
#include <math.h>

typedef float v2f __attribute__((ext_vector_type(2)));
typedef float v8f __attribute__((ext_vector_type(8)));

#define NDIM        128
#define ROWS_PER_WG 64
#define KSTEPS      32          // 128 / 4 (K per v_wmma_f32_16x16x4_f32)
#define ZSTRIDE     130         // padded LDS row stride: (2*lane + k) % 64 banks, conflict-free
#define MAX_ITER    500
#define EPS_F       1e-5f

// tanh via hardware transcendentals: tanh(y) = 1 - 2/(exp2(2*log2e*y) + 1).
// v_exp_f32 + v_rcp_f32 are ~1 ulp => |err| ~ 1e-7, far below the 1e-5 tolerance.
// Saturates correctly: y -> +inf: exp2->inf, rcp->0, result 1; y -> -inf: exp2->0, result -1.
__device__ __forceinline__ float fast_tanh(float y) {
    const float TWO_LOG2E = 2.885390081777927f;   // 2 * log2(e)
    float e = __builtin_amdgcn_exp2f(y * TWO_LOG2E);  // v_exp_f32
    float r = __builtin_amdgcn_rcpf(e + 1.0f);        // v_rcp_f32
    return __builtin_fmaf(-2.0f, r, 1.0f);
}

__global__ __launch_bounds__(256)
void tanh_fixed_point_kernel(const float* __restrict__ x,
                             const float* __restrict__ W,
                             float* __restrict__ out)
{
    __shared__ float    zlds[ROWS_PER_WG * ZSTRIDE];
    __shared__ unsigned res[2];

    const int tid     = threadIdx.x;
    const int lane    = tid & 31;
    const int wave    = tid >> 5;       // 0..7: owns output cols [16*wave, 16*wave+16)
    const int lo      = lane & 15;
    const int hi      = lane >> 4;      // selects K-half / M-half per WMMA layout
    const int rowbase = blockIdx.x * ROWS_PER_WG;

    // ---- one-time loads: B fragments (W^T tiles) pinned in VGPRs -------
    // B[k][n] = W[n][k]; lane lo = n - 16*wave; frag ks covers k = 4*ks + 2*hi, +1.
    v2f bfrag[KSTEPS];
    {
        const int j = wave * 16 + lo;   // output column == row of W
        #pragma unroll
        for (int ks = 0; ks < KSTEPS; ++ks) {
            const int k = 4 * ks + 2 * hi;
            bfrag[ks] = *(const v2f*)(W + j * NDIM + k);
        }
    }

    // x at this lane's D-matrix positions: 4 row tiles x 8 accumulator regs.
    float xreg[4][8];
    {
        const int col = wave * 16 + lo;
        #pragma unroll
        for (int rt = 0; rt < 4; ++rt)
            #pragma unroll
            for (int r = 0; r < 8; ++r) {
                const int row = rowbase + rt * 16 + r + 8 * hi;
                xreg[rt][r] = x[row * NDIM + col];
            }
    }

    // ---- init z = 0 (reference starts Newton from z0 = 0) ---------------
    for (int i = tid; i < ROWS_PER_WG * ZSTRIDE; i += 256) zlds[i] = 0.0f;
    if (tid == 0) { res[0] = 0u; res[1] = 0u; }
    __syncthreads();

    // ---- fixed-point iteration: z <- tanh(z @ W^T + x) -------------------
    for (int it = 0; it < MAX_ITER; ++it) {
        // Phase A: Y[:, wave cols] = z @ W^T via fp32 WMMA, 4 row tiles.
        v8f acc[4];
        #pragma unroll
        for (int rt = 0; rt < 4; ++rt) {
            v8f c = {};
            const float* zrow = zlds + (rt * 16 + lo) * ZSTRIDE;  // A-matrix row for this lane
            #pragma unroll
            for (int ks = 0; ks < KSTEPS; ++ks) {
                const int k = 4 * ks + 2 * hi;
                v2f a = *(const v2f*)(zrow + k);                  // ds_load_2addr_b64, conflict-free
                c = __builtin_amdgcn_wmma_f32_16x16x4_f32(
                        /*neg_a=*/false, a, /*neg_b=*/false, bfrag[ks],
                        /*c_mod=*/(short)0, c, /*reuse_a=*/false, /*reuse_b=*/false);
            }
            acc[rt] = c;
        }
        __syncthreads();    // all A-reads of zlds done before overwriting

        // Phase B: z_new = tanh(Y + x); write back; residual = max|z_new - z|.
        float m = 0.0f;
        const int col = wave * 16 + lo;
        #pragma unroll
        for (int rt = 0; rt < 4; ++rt) {
            #pragma unroll
            for (int r = 0; r < 8; ++r) {
                const int row = rt * 16 + r + 8 * hi;     // D layout: VGPR r -> M = r + 8*hi
                float zn = fast_tanh(acc[rt][r] + xreg[rt][r]);
                float zo = zlds[row * ZSTRIDE + col];
                m = fmaxf(m, fabsf(zn - zo));
                zlds[row * ZSTRIDE + col] = zn;
            }
        }
        // |f| as uint compares like float for non-negative values -> ds_max_u32
        atomicMax(&res[it & 1], __float_as_uint(m));
        __syncthreads();    // z writes + residual maxes visible

        const float gmax = __uint_as_float(res[it & 1]);
        if (tid == 0) res[(it + 1) & 1] = 0u;   // re-arm other slot (barrier-separated from its use)
        if (gmax < EPS_F) break;                // uniform across block: EXEC stays all-ones
    }
    __syncthreads();

    // ---- store converged z ----------------------------------------------
    for (int i = tid; i < ROWS_PER_WG * NDIM; i += 256) {
        const int r = i >> 7, c = i & (NDIM - 1);
        out[(rowbase + r) * NDIM + c] = zlds[r * ZSTRIDE + c];
    }
}

extern "C" void kernel_launch(void* const* d_in, const int* in_sizes, int n_in,
                              void* d_out, int out_size, void* d_ws, size_t ws_size,
                              hipStream_t stream) {
    (void)in_sizes; (void)n_in; (void)d_ws; (void)ws_size; (void)out_size;
    const float* x = (const float*)d_in[0];   // [4096, 128]
    const float* W = (const float*)d_in[1];   // [128, 128]
    float* out     = (float*)d_out;           // [4096, 128]

    const int batch  = 4096;
    const int blocks = batch / ROWS_PER_WG;   // 64 workgroups of 256 threads (8 waves)
    tanh_fixed_point_kernel<<<blocks, 256, 0, stream>>>(x, W, out);
}